// HybirdQuantumDCNN_3745211482334
// MI455X (gfx1250) — compile-verified
//
#include <hip/hip_runtime.h>
#include <math.h>

#define NQ   10
#define NL   5
#define NB   2048
#define NT   64
#define NF   128
#define NSTATE 1024
#define PI_F 3.14159265358979323846f
#define EPS_F 1e-5f

typedef __attribute__((ext_vector_type(2))) float v2f;
typedef __attribute__((ext_vector_type(8))) float v8f;

// ---------------------------------------------------------------- mean over T
// float4 (b128) loads: thread owns float4-column c of row b, strides over T.
__global__ void k_mean(const float4* __restrict__ x4, float4* __restrict__ xm4) {
    int tid = (int)threadIdx.x;                 // 256
    int b = blockIdx.x * 8 + (tid >> 5);        // 8 rows per block
    int c = tid & 31;                           // 32 float4-cols = 128 floats
    const float4* p = x4 + (size_t)b * NT * 32 + c;
    __builtin_prefetch(p + 32 * 32, 0, 1);      // global_prefetch_b8
    float sx = 0.f, sy = 0.f, sz = 0.f, sw = 0.f;
#pragma unroll 8
    for (int t = 0; t < NT; ++t) {
        float4 v = p[t * 32];
        sx += v.x; sy += v.y; sz += v.z; sw += v.w;
    }
    float4 o; o.x = sx * (1.0f / NT); o.y = sy * (1.0f / NT);
    o.z = sz * (1.0f / NT); o.w = sw * (1.0f / NT);
    xm4[b * 32 + c] = o;
}

// --------------------------------- Bpad[k*16+n] = (n<10) ? Wred[n*128+k] : 0
__global__ void k_bpad(const float* __restrict__ Wred, float* __restrict__ Bpad) {
    int idx = blockIdx.x * 256 + threadIdx.x;   // 2048 = 128*16
    int k = idx >> 4, n = idx & 15;
    Bpad[idx] = (n < NQ) ? Wred[n * NF + k] : 0.f;
}

// ------------------------------------------ feat = relu(xm @ Wred^T + bred), WMMA f32
// One wave per 16-row tile of M.  N padded 10 -> 16.  K = 128 in steps of 4.
__global__ void k_feat_wmma(const float* __restrict__ xm,
                            const float* __restrict__ Bpad,
                            const float* __restrict__ bred,
                            float* __restrict__ feat) {
    int wave  = (int)((blockIdx.x * blockDim.x + threadIdx.x) >> 5);
    int lane  = (int)(threadIdx.x & 31);
    int half  = lane >> 4;        // 0: lanes 0-15, 1: lanes 16-31
    int n16   = lane & 15;
    int row0  = wave * 16;

    const float* ap0 = xm + (row0 + n16) * NF + 2 * half;   // A: M=n16, K pair
    const float* bp0 = Bpad + half * 16 + n16;              // B: rows k0+half, k0+2+half

    v8f acc = {};
#pragma unroll
    for (int k0 = 0; k0 < NF; k0 += 4) {
        v2f a, bm;
        a.x  = ap0[k0];                 // K = k0 + 2*half
        a.y  = ap0[k0 + 1];             // K = k0 + 2*half + 1
        bm.x = bp0[k0 * 16];            // K = k0 + half
        bm.y = bp0[k0 * 16 + 32];       // K = k0 + half + 2
        acc = __builtin_amdgcn_wmma_f32_16x16x4_f32(false, a, false, bm,
                                                    (short)0, acc, false, false);
    }
    float bias = (n16 < NQ) ? bred[n16] : 0.f;
#pragma unroll
    for (int r = 0; r < 8; ++r) {
        // C/D layout: VGPR r -> M = r (lanes 0-15) or r+8 (lanes 16-31), N = lane&15
        int m = row0 + r + 8 * half;
        float v = acc[r] + bias;
        feat[m * 16 + n16] = v > 0.f ? v : 0.f;
    }
}

// ---------------------------------------------------------------- quantum circuit
__device__ __forceinline__ void pair_idx(int p, int bitpos, int& i0, int& i1) {
    int bit = 1 << bitpos;
    i0 = ((p >> bitpos) << (bitpos + 1)) | (p & (bit - 1));
    i1 = i0 | bit;
}

__device__ __forceinline__ void do_ry(float* sre, float* sim, int q, float th, int tid) {
    float c = __cosf(0.5f * th), s = __sinf(0.5f * th);
    int i0, i1; pair_idx(tid, 9 - q, i0, i1);
    float r0 = sre[i0], m0 = sim[i0], r1 = sre[i1], m1 = sim[i1];
    sre[i0] = c * r0 - s * r1;  sim[i0] = c * m0 - s * m1;
    sre[i1] = s * r0 + c * r1;  sim[i1] = s * m0 + c * m1;
    __syncthreads();
}

__device__ __forceinline__ void do_phase(float* sre, float* sim, int q, float phi, int tid) {
    float cp = __cosf(phi), sp = __sinf(phi);
    int i0, i1; pair_idx(tid, 9 - q, i0, i1);
    float r1 = sre[i1], m1 = sim[i1];
    sre[i1] = r1 * cp - m1 * sp;
    sim[i1] = r1 * sp + m1 * cp;
    __syncthreads();
}

__device__ __forceinline__ void do_cnot(float* sre, float* sim, int c, int t, int tid) {
    int cb = 1 << (9 - c), tb = 1 << (9 - t);
#pragma unroll
    for (int k = 0; k < 2; ++k) {
        int a = tid + k * 512;
        if ((a & cb) && !(a & tb)) {
            int a2 = a | tb;
            float r = sre[a], m = sim[a];
            sre[a] = sre[a2]; sim[a] = sim[a2];
            sre[a2] = r;      sim[a2] = m;
        }
    }
    __syncthreads();
}

__global__ void __launch_bounds__(512)
k_circuit(const float* __restrict__ feat, const float* __restrict__ qw,
          float* __restrict__ qout) {
    __shared__ float sre[NSTATE];
    __shared__ float sim[NSTATE];
    __shared__ float sang[NQ];
    __shared__ float sw[2 * NQ * NL];
    __shared__ float red[512];

    int b = blockIdx.x;
    int tid = (int)threadIdx.x;

    if (tid < NQ)          sang[tid] = feat[b * 16 + tid];
    if (tid < 2 * NQ * NL) sw[tid]   = qw[tid];
    // H^N |0> = uniform superposition, amplitude 2^{-NQ/2}
    sre[tid] = 0.03125f;  sre[tid + 512] = 0.03125f;
    sim[tid] = 0.f;       sim[tid + 512] = 0.f;
    __syncthreads();

    // temporal phase encoding
    for (int q = 0; q < NQ; ++q)
        do_phase(sre, sim, q, sang[q] * PI_F * (float)(2 * q + 1) / (float)NQ, tid);
    // RY amplitude encoding
    for (int q = 0; q < NQ; ++q)
        do_ry(sre, sim, q, sang[q] * PI_F, tid);

    for (int layer = 0; layer < NL; ++layer) {
        int ls  = layer * 2 * NQ;
        int dil = 1 << (2 * layer);
        int cnt = NQ - dil; if (cnt < 1) cnt = 1;
        for (int i = 0; i < cnt; ++i) {
            int t = i + dil; if (t > NQ - 1) t = NQ - 1;
            do_cnot(sre, sim, i, t, tid);
        }
        for (int i = 0; i < NQ; ++i) {
            float pf = __sinf((float)(i + 1) * (PI_F / (float)NQ));
            do_ry(sre, sim, i, sw[ls + i] * pf, tid);
            do_phase(sre, sim, i, sw[ls + NQ + i] * pf, tid);
        }
    }
    for (int i = 0; i < NQ - 1; ++i) do_cnot(sre, sim, i, i + 1, tid);

    // expectations: <Z0> (bit 9), <X1> (bit 8), <Y2> (bit 7)
    float ez = 0.f, ex = 0.f, ey = 0.f;
#pragma unroll
    for (int k = 0; k < 2; ++k) {
        int a = tid + k * 512;
        float r = sre[a], m = sim[a];
        float p = r * r + m * m;
        ez += (a & (1 << 9)) ? -p : p;
        int ax = a ^ (1 << 8);
        ex += r * sre[ax] + m * sim[ax];
        int b7 = 1 << 7;
        int a0 = a & ~b7, a1 = a | b7;
        ey += sre[a0] * sim[a1] - sim[a0] * sre[a1];
    }
    float vals[3] = {ez, ex, ey};
#pragma unroll
    for (int j = 0; j < 3; ++j) {
        red[tid] = vals[j];
        __syncthreads();
        for (int s = 256; s > 0; s >>= 1) {
            if (tid < s) red[tid] += red[tid + s];
            __syncthreads();
        }
        if (tid == 0) qout[b * 3 + j] = red[0];
        __syncthreads();
    }
}

// ---------------------------------------------------------------- MLP head
__global__ void k_h1(const float* __restrict__ q, const float* __restrict__ W1,
                     const float* __restrict__ b1, float* __restrict__ h1) {
    int idx = blockIdx.x * 256 + threadIdx.x;   // 2048*32
    int b = idx >> 5, j = idx & 31;
    float acc = b1[j];
#pragma unroll
    for (int k = 0; k < 3; ++k) acc += q[b * 3 + k] * W1[j * 3 + k];
    h1[idx] = acc;
}

__global__ void k_stats(const float* __restrict__ h, int ncols, float* __restrict__ st) {
    __shared__ float ss[256], s2[256];
    int j = blockIdx.x, tid = (int)threadIdx.x;
    float a = 0.f, b = 0.f;
    for (int r = tid; r < NB; r += 256) {
        float v = h[r * ncols + j];
        a += v; b += v * v;
    }
    ss[tid] = a; s2[tid] = b;
    __syncthreads();
    for (int k = 128; k > 0; k >>= 1) {
        if (tid < k) { ss[tid] += ss[tid + k]; s2[tid] += s2[tid + k]; }
        __syncthreads();
    }
    if (tid == 0) {
        float m = ss[0] * (1.0f / NB);
        st[j] = m;
        st[ncols + j] = s2[0] * (1.0f / NB) - m * m;   // biased variance
    }
}

__global__ void k_h2(const float* __restrict__ h1, const float* __restrict__ st1,
                     const float* __restrict__ g1, const float* __restrict__ be1,
                     const float* __restrict__ W2, const float* __restrict__ b2,
                     float* __restrict__ h2) {
    int idx = blockIdx.x * 256 + threadIdx.x;   // 2048*16
    int b = idx >> 4, i = idx & 15;
    float acc = b2[i];
#pragma unroll
    for (int j = 0; j < 32; ++j) {
        float v = h1[b * 32 + j];
        v = g1[j] * (v - st1[j]) * rsqrtf(st1[32 + j] + EPS_F) + be1[j];
        v = v > 0.f ? v : 0.f;
        acc += v * W2[i * 32 + j];
    }
    h2[idx] = acc;
}

__global__ void k_out(const float* __restrict__ h2, const float* __restrict__ st2,
                      const float* __restrict__ g2, const float* __restrict__ be2,
                      const float* __restrict__ W3, const float* __restrict__ b3,
                      float* __restrict__ out) {
    int b = blockIdx.x * 256 + threadIdx.x;     // 2048
    float acc = b3[0];
#pragma unroll
    for (int i = 0; i < 16; ++i) {
        float v = h2[b * 16 + i];
        v = g2[i] * (v - st2[i]) * rsqrtf(st2[16 + i] + EPS_F) + be2[i];
        v = v > 0.f ? v : 0.f;
        acc += v * W3[i];
    }
    out[b] = acc;
}

// ---------------------------------------------------------------- launch
extern "C" void kernel_launch(void* const* d_in, const int* in_sizes, int n_in,
                              void* d_out, int out_size, void* d_ws, size_t ws_size,
                              hipStream_t stream) {
    const float* x    = (const float*)d_in[0];
    const float* Wred = (const float*)d_in[1];
    const float* bred = (const float*)d_in[2];
    const float* qw   = (const float*)d_in[3];
    const float* W1   = (const float*)d_in[4];
    const float* b1   = (const float*)d_in[5];
    const float* g1   = (const float*)d_in[6];
    const float* be1  = (const float*)d_in[7];
    const float* W2   = (const float*)d_in[8];
    const float* b2   = (const float*)d_in[9];
    const float* g2   = (const float*)d_in[10];
    const float* be2  = (const float*)d_in[11];
    const float* W3   = (const float*)d_in[12];
    const float* b3   = (const float*)d_in[13];
    float* out = (float*)d_out;

    float* xm   = (float*)d_ws;            // 2048*128
    float* Bpad = xm   + NB * NF;          // 128*16
    float* feat = Bpad + NF * 16;          // 2048*16
    float* qo   = feat + NB * 16;          // 2048*3
    float* h1   = qo   + NB * 3;           // 2048*32
    float* st1  = h1   + NB * 32;          // 64
    float* h2   = st1  + 64;               // 2048*16
    float* st2  = h2   + NB * 16;          // 32

    k_mean     <<<NB / 8, 256, 0, stream>>>((const float4*)x, (float4*)xm);
    k_bpad     <<<(NF * 16) / 256, 256, 0, stream>>>(Wred, Bpad);
    k_feat_wmma<<<NB / 16 / 4, 128, 0, stream>>>(xm, Bpad, bred, feat);
    k_circuit  <<<NB, 512, 0, stream>>>(feat, qw, qo);
    k_h1       <<<NB * 32 / 256, 256, 0, stream>>>(qo, W1, b1, h1);
    k_stats    <<<32, 256, 0, stream>>>(h1, 32, st1);
    k_h2       <<<NB * 16 / 256, 256, 0, stream>>>(h1, st1, g1, be1, W2, b2, h2);
    k_stats    <<<16, 256, 0, stream>>>(h2, 16, st2);
    k_out      <<<NB / 256, 256, 0, stream>>>(h2, st2, g2, be2, W3, b3, out);
}